// STDPMNIST_4999341932697
// MI455X (gfx1250) — compile-verified
//
#include <hip/hip_runtime.h>

// ---------------- problem constants ----------------
#define TT     15
#define CIN    2
#define HW     384
#define PLANE  (HW*HW)          // 147456
#define COUT   30
#define KSZ    5
#define KRED   50               // CIN*K*K
#define KPAD   52               // padded to multiple of 4 for wmma f32 16x16x4
#define THRESH 15.0f
#define KWTA   5
#define RAD    2
#define XTILE  128              // x positions per block (3 tiles cover 384)
#define NPART  288              // partial-argmax blocks

typedef float v2f __attribute__((ext_vector_type(2)));
typedef float v8f __attribute__((ext_vector_type(8)));

// compile-time im2col offset for reduction index k = (c*25 + ky*5 + kx)
// into s_in laid out as [c][ky][132-wide x halo]
#define AOFF(k) (((k) / 25) * 660 + ((((k) % 25) / 5) * 132) + ((k) % 5))

// =====================================================================
// K1: conv(5x5, pad2) + threshold-fire via V_WMMA_F32_16X16X4_F32.
// GEMM view: M = 16 x-positions, N = 16 channels (x2 groups), K = 50->52.
// One block = 8 wave32s = one (t, y, 128-wide x tile); pot written to d_out.
// All per-step LDS offsets are compile-time constants selected by the lane's
// K-half bit (hi), so the inner loop is selects + ds loads + wmma only.
// =====================================================================
__global__ __launch_bounds__(256) void k_conv_fire(const float* __restrict__ in,
                                                   const float* __restrict__ w1,
                                                   float* __restrict__ pot)
{
    __shared__ float s_in[CIN * 5 * 132];   // [c][ky][x halo], 5.28 KB
    __shared__ float s_w [KPAD * 32];       // [k][ch], zero padded, 6.5 KB
    __shared__ float s_tile[32 * 132];      // transpose staging, 16.5 KB

    const int tid  = threadIdx.x;
    const int bid  = blockIdx.x;
    const int t    = bid / (HW * 3);
    const int rem  = bid % (HW * 3);
    const int y    = rem / 3;
    const int xb   = (rem % 3) * XTILE;

    // stage input halo rows y-2..y+2, x in [xb-2, xb+130), zero-pad OOB
    for (int i = tid; i < CIN * 5 * 132; i += 256) {
        int c  = i / 660;
        int r2 = i % 660;
        int ry = r2 / 132;
        int rx = r2 % 132;
        int gy = y + ry - 2;
        int gx = xb + rx - 2;
        float v = 0.0f;
        if (gy >= 0 && gy < HW && gx >= 0 && gx < HW)
            v = in[((t * CIN + c) * HW + gy) * HW + gx];
        s_in[i] = v;
    }
    // stage weights, zero pad k>=50 and ch>=30.  w1 layout (COUT,CIN,5,5):
    // flat = ch*50 + (c*25 + ky*5 + kx) == ch*50 + k  (matches our k order)
    for (int i = tid; i < KPAD * 32; i += 256) {
        int k  = i >> 5;
        int ch = i & 31;
        float v = 0.0f;
        if (k < KRED && ch < COUT) v = w1[ch * KRED + k];
        s_w[i] = v;
    }
    __syncthreads();

    const int lane = tid & 31;
    const int wave = tid >> 5;
    const int m    = lane & 15;        // A-frag row / B,D column (channel)
    const bool hi  = (lane >> 4) != 0; // K sub-pair selector per ISA layout
    const int xloc = wave * 16;        // wave's x sub-tile inside block

    v8f acc0 = {};                     // channels 0..15
    v8f acc1 = {};                     // channels 16..31 (30,31 padded zero)

    const float* a_base = s_in + xloc + m;  // patch row for this lane's M
    const float* b_base = s_w + m;          // channel column for this lane's N

    #pragma unroll
    for (int s = 0; s < KPAD / 4; ++s) {
        // lanes 0-15 carry K = 4s,4s+1 ; lanes 16-31 carry K = 4s+2,4s+3.
        // Both alternatives are compile-time constants -> single v_cndmask each.
        const int offA0 = hi ? AOFF(4 * s + 2) : AOFF(4 * s);
        const int offA1 = hi ? AOFF(4 * s + 3) : AOFF(4 * s + 1);
        const int offB  = (hi ? (4 * s + 2) : (4 * s)) * 32;

        float a0 = a_base[offA0];
        float a1 = a_base[offA1];
        if (s == KPAD / 4 - 1) {       // only K=50,51 (s=12, hi lanes) are pad
            if (hi) { a0 = 0.0f; a1 = 0.0f; }
        }
        v2f a  = { a0, a1 };
        v2f b0 = { b_base[offB],      b_base[offB + 32]      };
        v2f b1 = { b_base[offB + 16], b_base[offB + 48]      };
        acc0 = __builtin_amdgcn_wmma_f32_16x16x4_f32(false, a, false, b0,
                                                     (short)0, acc0, false, false);
        acc1 = __builtin_amdgcn_wmma_f32_16x16x4_f32(false, a, false, b1,
                                                     (short)0, acc1, false, false);
    }

    // D layout: VGPR r -> M = r + 8*hi (x position), N = lane&15 (channel)
    const int mrow = xloc + (hi ? 8 : 0);
    #pragma unroll
    for (int r = 0; r < 8; ++r) {
        const int mx = mrow + r;
        s_tile[m * 132 + mx]        = acc0[r];
        s_tile[(16 + m) * 132 + mx] = acc1[r];
    }
    __syncthreads();

    // threshold-fire + coalesced store into pot half of d_out
    for (int i = tid; i < COUT * XTILE; i += 256) {
        int ch = i / XTILE;
        int x  = i & (XTILE - 1);
        float v = s_tile[ch * 132 + x];
        pot[((t * COUT + ch) * HW + y) * HW + xb + x] = (v > THRESH) ? v : 0.0f;
    }
}

// =====================================================================
// K0: tiny init (global max accumulator for the k-winner offset "v")
// =====================================================================
__global__ void k_init(unsigned* vmax)
{
    if (threadIdx.x == 0 && blockIdx.x == 0) *vmax = 0u;
}

// =====================================================================
// K2: pointwise inhibition (per pixel across T and F), in-place on pot,
// writes spk, and per-pixel (winner feature, nspikes, first-spike value).
// Consecutive threads = consecutive w => every access coalesced.
// =====================================================================
__global__ __launch_bounds__(256) void k_inhibit(float* __restrict__ pot,
                                                 float* __restrict__ spk,
                                                 int*   __restrict__ winmap,
                                                 int*   __restrict__ nspmap,
                                                 float* __restrict__ valmap,
                                                 unsigned* __restrict__ vmax)
{
    const int pos = blockIdx.x * 256 + threadIdx.x;
    if (pos >= PLANE) return;

    // pass A: per-t any-fire + last-step fired flag
    int cnt = 0;
    float mv_last = 0.0f;
    #pragma unroll 1
    for (int t = 0; t < TT; ++t) {
        float mv = 0.0f;
        #pragma unroll
        for (int f = 0; f < COUT; ++f)
            mv = fmaxf(mv, pot[(t * COUT + f) * PLANE + pos]);
        cnt += (mv > 0.0f) ? 1 : 0;
        if (t == TT - 1) mv_last = mv;
    }
    const bool fired = (mv_last > 0.0f);
    int earliest = TT - cnt;
    earliest = earliest < 0 ? 0 : (earliest > TT - 1 ? TT - 1 : earliest);

    // pass B: winning feature = argmax over F at earliest (ties -> lowest f)
    float bw = 0.0f;
    int wi = 0;
    #pragma unroll
    for (int f = 0; f < COUT; ++f) {
        float p = pot[(earliest * COUT + f) * PLANE + pos];
        if (p > bw) { bw = p; wi = f; }
    }

    // pass C: winner stats for get_k_winners (post-inhibition view)
    int nsp = 0;
    #pragma unroll 1
    for (int t = 0; t < TT; ++t) {
        float p = fired ? pot[(t * COUT + wi) * PLANE + pos] : 0.0f;
        nsp += (p > 0.0f) ? 1 : 0;
    }
    int ef = TT - nsp;
    ef = ef < 0 ? 0 : (ef > TT - 1 ? TT - 1 : ef);
    float val = (fired && nsp > 0) ? pot[(ef * COUT + wi) * PLANE + pos] : 0.0f;

    // pass D: apply coef (zero losers / non-fired), write spikes
    #pragma unroll 1
    for (int t = 0; t < TT; ++t) {
        #pragma unroll
        for (int f = 0; f < COUT; ++f) {
            const int idx = (t * COUT + f) * PLANE + pos;
            if (f == wi && fired) {
                float p = pot[idx];                 // keep pot as-is
                spk[idx] = (p > 0.0f) ? 1.0f : 0.0f;
            } else {
                pot[idx] = 0.0f;
                spk[idx] = 0.0f;
            }
        }
    }

    winmap[pos] = fired ? wi : -1;
    nspmap[pos] = nsp;
    valmap[pos] = val;
    if (nsp > 0 && val > 0.0f)
        atomicMax(vmax, __float_as_uint(val));       // val >= 0 => uint-order ok
}

// =====================================================================
// K3a: partial argmax over candidate pixels with suppression predicates
// total[pixel] = nsp * (val + vmax*T), flat idx = f*PLANE + pos (tie: min idx)
// =====================================================================
__global__ __launch_bounds__(256) void k_argmax_part(const int* __restrict__ winmap,
                                                     const int* __restrict__ nspmap,
                                                     const float* __restrict__ valmap,
                                                     const unsigned* __restrict__ vmax,
                                                     const int* __restrict__ wtmp,
                                                     int iter,
                                                     float* __restrict__ pval,
                                                     int* __restrict__ pidx)
{
    const float voff = __uint_as_float(*vmax) * (float)TT;
    float best = -1.0f;
    int bidx = 0;

    for (int pos = blockIdx.x * 256 + threadIdx.x; pos < PLANE; pos += gridDim.x * 256) {
        const int wf = winmap[pos];
        if (wf < 0) continue;
        const int h = pos / HW, w = pos % HW;
        bool skip = false;
        for (int j = 0; j < iter; ++j) {
            const int fj = wtmp[j * 4 + 0];
            const int rj = wtmp[j * 4 + 1];
            const int cj = wtmp[j * 4 + 2];
            if (wf == fj) skip = true;                 // whole-feature suppression
            int dr = h - rj; dr = dr < 0 ? -dr : dr;   // columnar window suppression
            int dc = w - cj; dc = dc < 0 ? -dc : dc;
            if (dr <= RAD && dc <= RAD) skip = true;
        }
        if (skip) continue;
        const float tot = (float)nspmap[pos] * (valmap[pos] + voff);
        const int fidx = wf * PLANE + pos;
        if (tot > best || (tot == best && fidx < bidx)) { best = tot; bidx = fidx; }
    }

    __shared__ float sv[256];
    __shared__ int   si[256];
    sv[threadIdx.x] = best;
    si[threadIdx.x] = bidx;
    __syncthreads();
    for (int off = 128; off > 0; off >>= 1) {
        if (threadIdx.x < off) {
            float ov = sv[threadIdx.x + off];
            int   oi = si[threadIdx.x + off];
            if (ov > sv[threadIdx.x] || (ov == sv[threadIdx.x] && oi < si[threadIdx.x])) {
                sv[threadIdx.x] = ov; si[threadIdx.x] = oi;
            }
        }
        __syncthreads();
    }
    if (threadIdx.x == 0) { pval[blockIdx.x] = sv[0]; pidx[blockIdx.x] = si[0]; }
}

// =====================================================================
// K3b: final reduce of NPART partials; record winner (and -1s if invalid)
// =====================================================================
__global__ __launch_bounds__(256) void k_argmax_final(const float* __restrict__ pval,
                                                      const int* __restrict__ pidx,
                                                      int iter,
                                                      int* __restrict__ wtmp,
                                                      float* __restrict__ wout)
{
    float best = -1.0f;
    int bidx = 0;
    for (int i = threadIdx.x; i < NPART; i += 256) {
        float v = pval[i];
        int id = pidx[i];
        if (v > best || (v == best && id < bidx)) { best = v; bidx = id; }
    }
    __shared__ float sv[256];
    __shared__ int   si[256];
    sv[threadIdx.x] = best;
    si[threadIdx.x] = bidx;
    __syncthreads();
    for (int off = 128; off > 0; off >>= 1) {
        if (threadIdx.x < off) {
            float ov = sv[threadIdx.x + off];
            int   oi = si[threadIdx.x + off];
            if (ov > sv[threadIdx.x] || (ov == sv[threadIdx.x] && oi < si[threadIdx.x])) {
                sv[threadIdx.x] = ov; si[threadIdx.x] = oi;
            }
        }
        __syncthreads();
    }
    if (threadIdx.x == 0) {
        const int f = si[0] / PLANE;
        const int rem = si[0] % PLANE;
        const int r = rem / HW;
        const int c = rem % HW;
        const bool valid = sv[0] > 0.0f;
        // suppression coords recorded regardless of validity (matches reference)
        wtmp[iter * 4 + 0] = f;
        wtmp[iter * 4 + 1] = r;
        wtmp[iter * 4 + 2] = c;
        wtmp[iter * 4 + 3] = valid ? 1 : 0;
        wout[iter * 3 + 0] = valid ? (float)f : -1.0f;
        wout[iter * 3 + 1] = valid ? (float)r : -1.0f;
        wout[iter * 3 + 2] = valid ? (float)c : -1.0f;
    }
}

// =====================================================================
extern "C" void kernel_launch(void* const* d_in, const int* in_sizes, int n_in,
                              void* d_out, int out_size, void* d_ws, size_t ws_size,
                              hipStream_t stream)
{
    const float* in = (const float*)d_in[0];   // (15,2,384,384)
    const float* w1 = (const float*)d_in[1];   // (30,2,5,5)
    // d_in[2] = layer_idx (==1), unused

    const int TOT = TT * COUT * PLANE;         // 66,355,200
    float* spk = (float*)d_out;                // output 0
    float* pot = (float*)d_out + TOT;          // output 1 (also conv staging)
    float* wout = (float*)d_out + 2 * TOT;     // output 2: winners (5,3) as float

    // workspace layout (bytes)
    char* ws = (char*)d_ws;
    unsigned* vmax = (unsigned*)(ws + 0);
    int*   wtmp    = (int*)  (ws + 64);                    // 5*4 ints
    float* pval    = (float*)(ws + 256);                   // NPART floats
    int*   pidx    = (int*)  (ws + 256 + NPART * 4);       // NPART ints
    int*   winmap  = (int*)  (ws + 8192);                  // PLANE ints
    float* valmap  = (float*)(ws + 8192 + PLANE * 4);      // PLANE floats
    int*   nspmap  = (int*)  (ws + 8192 + 2 * PLANE * 4);  // PLANE ints
    (void)ws_size; (void)in_sizes; (void)n_in; (void)out_size;

    // 1) conv + threshold fire (WMMA f32) -> pot
    k_conv_fire<<<dim3(TT * HW * 3), dim3(256), 0, stream>>>(in, w1, pot);

    // 2) init global max for the k-winner offset
    k_init<<<dim3(1), dim3(64), 0, stream>>>(vmax);

    // 3) pointwise inhibition -> pot (in place), spk, per-pixel winner maps
    k_inhibit<<<dim3(PLANE / 256), dim3(256), 0, stream>>>(pot, spk, winmap,
                                                           nspmap, valmap, vmax);

    // 4) five sequential k-winner selections with suppression
    for (int it = 0; it < KWTA; ++it) {
        k_argmax_part<<<dim3(NPART), dim3(256), 0, stream>>>(winmap, nspmap, valmap,
                                                             vmax, wtmp, it, pval, pidx);
        k_argmax_final<<<dim3(1), dim3(256), 0, stream>>>(pval, pidx, it, wtmp, wout);
    }
}